// STGCN_40200893890746
// MI455X (gfx1250) — compile-verified
//
#include <hip/hip_runtime.h>
#include <hip/hip_bf16.h>
#include <stdint.h>

#define N_NODES 50000
#define F_INC   128
#define HID     128
#define OUT_C   64
#define NLAYERS 3
#define KT      9
#define PADT    4
#define NEDGES  600000
#define BN_EPS  1e-5f

typedef __attribute__((ext_vector_type(2))) float v2f;
typedef __attribute__((ext_vector_type(8))) float v8f;
typedef int v4i128 __attribute__((vector_size(16)));   // matches builtin's b128 pointee

#define AS1 __attribute__((address_space(1)))
#define AS3 __attribute__((address_space(3)))

#if defined(__AMDGCN__) && __has_builtin(__builtin_amdgcn_global_load_async_to_lds_b128) && __has_builtin(__builtin_amdgcn_s_wait_asynccnt)
#define HAVE_ASYNC_LDS 1
#else
#define HAVE_ASYNC_LDS 0
#endif

// ---------------- degree / inverse degree ----------------
__global__ void deg_kernel(const int* __restrict__ ei, float* __restrict__ deg) {
    int e = blockIdx.x * blockDim.x + threadIdx.x;
    if (e < NEDGES) {
        int dst = ei[NEDGES + e];
        atomicAdd(&deg[dst], 1.0f);
    }
}

__global__ void invdeg_kernel(const float* __restrict__ deg, float* __restrict__ invd) {
    int n = blockIdx.x * blockDim.x + threadIdx.x;
    if (n < N_NODES) {
        float d = deg[n];
        invd[n] = d > 0.0f ? 1.0f / fmaxf(d, 1.0f) : 0.0f;
    }
}

// ---------------- weight re-layouts ----------------
// Pair-interleaved B layout so one b64 per lane fetches both K values:
// wt[i][s][c/2][o][c&1] = tw[i][o][c][s]
__global__ void prep_conv_w(const float* __restrict__ tw, float* __restrict__ wt) {
    int idx = blockIdx.x * blockDim.x + threadIdx.x;
    const int total = NLAYERS * KT * HID * HID;
    if (idx >= total) return;
    int o    = idx & 127;
    int c    = (idx >> 7) & 127;
    int rest = idx >> 14;          // i*9 + s   (HID*HID = 2^14)
    int s    = rest % KT;
    int i    = rest / KT;
    size_t dst = (size_t)i * KT * HID * HID
               + ((((size_t)(s * 64 + (c >> 1))) * HID + o) << 1) + (c & 1);
    wt[dst] = tw[(((size_t)i * HID + o) * F_INC + c) * KT + s];
}

// wcat[i][k/2][o][k&1]: k<128 -> wl[i][o][k]; k>=128 -> wr[i][o][k-128]
__global__ void prep_cat_w(const float* __restrict__ wl, const float* __restrict__ wr,
                           float* __restrict__ wcat) {
    int idx = blockIdx.x * blockDim.x + threadIdx.x;
    const int total = NLAYERS * 256 * HID;
    if (idx >= total) return;
    int o = idx & 127;
    int k = (idx >> 7) & 255;
    int i = idx >> 15;             // 256*128 = 2^15
    float v;
    if (k < 128) v = wl[((size_t)i * HID + o) * HID + k];
    else         v = wr[((size_t)i * HID + o) * HID + (k - 128)];
    size_t dst = (size_t)i * 256 * HID + ((((size_t)(k >> 1)) * HID + o) << 1) + (k & 1);
    wcat[dst] = v;
}

// ---------------- temporal conv (im2col-free shifted WMMA GEMM) ----------------
// block: 256 threads = 8 waves; 64 nodes per block (4 M-tiles), wave w -> cout [16w,16w+16)
__global__ __launch_bounds__(256)
void conv_wmma_kernel(const float* __restrict__ x, const float* __restrict__ wt,
                      const float* __restrict__ bias, float* __restrict__ h) {
    __shared__ float lds[72 * 128];          // 36 KB halo'd slab
    const int m0  = blockIdx.x * 64;
    const int tid = threadIdx.x;

    // cooperative fill: 72 rows x 128 cols, zero padded at node-axis boundaries
    for (int f = tid; f < 72 * 32; f += 256) {   // b128 granules, 9 per thread (uniform trip)
        int r = f >> 5, c4 = f & 31;
        int node = m0 - PADT + r;
        float* ldsp = &lds[r * 128 + c4 * 4];
        if (node >= 0 && node < N_NODES) {
#if HAVE_ASYNC_LDS
            __builtin_amdgcn_global_load_async_to_lds_b128(
                (AS1 v4i128*)(uintptr_t)(x + (size_t)node * HID + c4 * 4),
                (AS3 v4i128*)ldsp, 0, 0);
#else
            *reinterpret_cast<float4*>(ldsp) =
                *reinterpret_cast<const float4*>(x + (size_t)node * HID + c4 * 4);
#endif
        } else {
            *reinterpret_cast<float4*>(ldsp) = make_float4(0.f, 0.f, 0.f, 0.f);
        }
    }
#if HAVE_ASYNC_LDS
    __builtin_amdgcn_s_wait_asynccnt(0);
#endif
    __syncthreads();

    const int wave = tid >> 5;
    const int lane = tid & 31;
    const int half = lane >> 4;     // A/B K-pair select per ISA 16x4 layout
    const int mrow = lane & 15;
    const int n0   = wave * 16;
    const int o    = n0 + mrow;     // this lane's output channel (B: N = lane%16)

    v8f acc0 = {0,0,0,0,0,0,0,0}, acc1 = {0,0,0,0,0,0,0,0};
    v8f acc2 = {0,0,0,0,0,0,0,0}, acc3 = {0,0,0,0,0,0,0,0};

    for (int s = 0; s < KT; ++s) {
        const float* wts = wt + s * 64 * 128 * 2;        // [c2][o][2]
        for (int c0 = 0; c0 < 128; c0 += 4) {
            const int cb = c0 + half * 2;
            // single coalesced b64: both K values for this lane's output channel
            v2f b = *reinterpret_cast<const v2f*>(wts + (((cb >> 1) * HID + o) << 1));
            const float* arow = &lds[(mrow + s) * 128 + cb];
            v2f a0 = *reinterpret_cast<const v2f*>(arow + 0 * 16 * 128);
            v2f a1 = *reinterpret_cast<const v2f*>(arow + 1 * 16 * 128);
            v2f a2 = *reinterpret_cast<const v2f*>(arow + 2 * 16 * 128);
            v2f a3 = *reinterpret_cast<const v2f*>(arow + 3 * 16 * 128);
            acc0 = __builtin_amdgcn_wmma_f32_16x16x4_f32(false, a0, false, b, (short)0, acc0, false, false);
            acc1 = __builtin_amdgcn_wmma_f32_16x16x4_f32(false, a1, false, b, (short)0, acc1, false, false);
            acc2 = __builtin_amdgcn_wmma_f32_16x16x4_f32(false, a2, false, b, (short)0, acc2, false, false);
            acc3 = __builtin_amdgcn_wmma_f32_16x16x4_f32(false, a3, false, b, (short)0, acc3, false, false);
        }
    }

    const float bv = bias[o];
    v8f accs[4] = {acc0, acc1, acc2, acc3};
#pragma unroll
    for (int mt = 0; mt < 4; ++mt) {
#pragma unroll
        for (int r = 0; r < 8; ++r) {           // C/D: M = r + 8*half, N = lane%16
            int node = m0 + mt * 16 + half * 8 + r;
            if (node < N_NODES) {
                float v = accs[mt][r] + bv;
                h[(size_t)node * HID + o] = fmaxf(v, 0.0f);
            }
        }
    }
}

// ---------------- SAGE mean-aggregation scatter ----------------
__global__ void scatter_kernel(const float* __restrict__ h, const int* __restrict__ ei,
                               float* __restrict__ agg) {
    long long t = (long long)blockIdx.x * blockDim.x + threadIdx.x;
    if (t >= (long long)NEDGES * 32) return;
    int e    = (int)(t >> 5);
    int lane = (int)(t & 31);
    int src = ei[e];
    int dst = ei[NEDGES + e];
    float4 v = *reinterpret_cast<const float4*>(h + (size_t)src * HID + lane * 4);
    float* base = agg + (size_t)dst * HID + lane * 4;
    atomicAdd(base + 0, v.x);
    atomicAdd(base + 1, v.y);
    atomicAdd(base + 2, v.z);
    atomicAdd(base + 3, v.w);
}

// ---------------- SAGE GEMM: x' = relu(bn((agg*invd)@wlT + h@wrT + bl)) ----------------
// block: 256 threads = 8 waves; 32 nodes per block (2 M-tiles); K = 256 (concat)
__global__ __launch_bounds__(256)
void sage_wmma_kernel(const float* __restrict__ agg, const float* __restrict__ invd,
                      const float* __restrict__ h, const float* __restrict__ wcat,
                      const float* __restrict__ bl,
                      const float* __restrict__ bng, const float* __restrict__ bnb,
                      const float* __restrict__ bnm, const float* __restrict__ bnv,
                      float* __restrict__ xout) {
    __shared__ float lds[32 * 256];          // 32 KB: A = [agg*invd | h]
    const int m0  = blockIdx.x * 32;
    const int tid = threadIdx.x;

    for (int f = tid; f < 32 * 64; f += 256) {   // b128 granules, 8 per thread (uniform trip)
        int r = f >> 6, col = (f & 63) * 4;
        int node = m0 + r;
        float* ldsp = &lds[r * 256 + col];
        if (node < N_NODES) {
            if (col < 128) {
                // needs inv_deg scaling -> stays on the VGPR path
                float s = invd[node];
                float4 t4 = *reinterpret_cast<const float4*>(agg + (size_t)node * HID + col);
                *reinterpret_cast<float4*>(ldsp) =
                    make_float4(t4.x * s, t4.y * s, t4.z * s, t4.w * s);
            } else {
#if HAVE_ASYNC_LDS
                __builtin_amdgcn_global_load_async_to_lds_b128(
                    (AS1 v4i128*)(uintptr_t)(h + (size_t)node * HID + (col - 128)),
                    (AS3 v4i128*)ldsp, 0, 0);
#else
                *reinterpret_cast<float4*>(ldsp) =
                    *reinterpret_cast<const float4*>(h + (size_t)node * HID + (col - 128));
#endif
            }
        } else {
            *reinterpret_cast<float4*>(ldsp) = make_float4(0.f, 0.f, 0.f, 0.f);
        }
    }
#if HAVE_ASYNC_LDS
    __builtin_amdgcn_s_wait_asynccnt(0);
#endif
    __syncthreads();

    const int wave = tid >> 5;
    const int lane = tid & 31;
    const int half = lane >> 4;
    const int mrow = lane & 15;
    const int n0   = wave * 16;
    const int o    = n0 + mrow;

    v8f acc0 = {0,0,0,0,0,0,0,0}, acc1 = {0,0,0,0,0,0,0,0};
    for (int k0 = 0; k0 < 256; k0 += 4) {
        const int kb = k0 + half * 2;
        v2f b = *reinterpret_cast<const v2f*>(wcat + (((kb >> 1) * HID + o) << 1));
        v2f a0 = *reinterpret_cast<const v2f*>(&lds[mrow * 256 + kb]);
        v2f a1 = *reinterpret_cast<const v2f*>(&lds[(mrow + 16) * 256 + kb]);
        acc0 = __builtin_amdgcn_wmma_f32_16x16x4_f32(false, a0, false, b, (short)0, acc0, false, false);
        acc1 = __builtin_amdgcn_wmma_f32_16x16x4_f32(false, a1, false, b, (short)0, acc1, false, false);
    }

    const float bias  = bl[o];
    const float scale = bng[o] * rsqrtf(bnv[o] + BN_EPS);
    const float mm = bnm[o], bb = bnb[o];
    v8f accs[2] = {acc0, acc1};
#pragma unroll
    for (int mt = 0; mt < 2; ++mt) {
#pragma unroll
        for (int r = 0; r < 8; ++r) {
            int node = m0 + mt * 16 + half * 8 + r;
            if (node < N_NODES) {
                float y = accs[mt][r] + bias;
                y = scale * (y - mm) + bb;
                xout[(size_t)node * HID + o] = fmaxf(y, 0.0f);
            }
        }
    }
}

// ---------------- classifier + log_softmax ----------------
__global__ __launch_bounds__(64)
void clf_kernel(const float* __restrict__ x, const float* __restrict__ w,
                const float* __restrict__ b, float* __restrict__ out) {
    __shared__ float xr[128];
    __shared__ float lg[64];
    const int node = blockIdx.x;
    const int t = threadIdx.x;
    xr[t]      = x[(size_t)node * HID + t];
    xr[t + 64] = x[(size_t)node * HID + t + 64];
    __syncthreads();
    float acc = b[t];
    const float* wrow = w + t * HID;
#pragma unroll 8
    for (int k = 0; k < HID; ++k) acc += xr[k] * wrow[k];
    lg[t] = acc;
    __syncthreads();
    float mx = -1e30f;
    for (int j = 0; j < OUT_C; ++j) mx = fmaxf(mx, lg[j]);
    float sum = 0.f;
    for (int j = 0; j < OUT_C; ++j) sum += expf(lg[j] - mx);
    out[(size_t)node * OUT_C + t] = acc - mx - logf(sum);
}

// ---------------- launcher ----------------
extern "C" void kernel_launch(void* const* d_in, const int* in_sizes, int n_in,
                              void* d_out, int out_size, void* d_ws, size_t ws_size,
                              hipStream_t stream) {
    (void)in_sizes; (void)n_in; (void)out_size; (void)ws_size;
    const float* x_in = (const float*)d_in[0];
    const int*   ei   = (const int*)d_in[1];
    const float* tw   = (const float*)d_in[2];
    const float* tb   = (const float*)d_in[3];
    const float* wl   = (const float*)d_in[4];
    const float* bl   = (const float*)d_in[5];
    const float* wr   = (const float*)d_in[6];
    const float* bng  = (const float*)d_in[7];
    const float* bnb  = (const float*)d_in[8];
    const float* bnm  = (const float*)d_in[9];
    const float* bnv  = (const float*)d_in[10];
    const float* clfw = (const float*)d_in[11];
    const float* clfb = (const float*)d_in[12];
    float* out = (float*)d_out;

    float* ws = (float*)d_ws;
    size_t off = 0;
    float* xbuf    = ws + off; off += (size_t)N_NODES * HID;
    float* hbuf    = ws + off; off += (size_t)N_NODES * HID;
    float* aggbuf  = ws + off; off += (size_t)N_NODES * HID;
    float* degbuf  = ws + off; off += N_NODES;
    float* invdbuf = ws + off; off += N_NODES;
    float* wtconv  = ws + off; off += (size_t)NLAYERS * KT * HID * HID;
    float* wcat    = ws + off; off += (size_t)NLAYERS * 256 * HID;

    (void)hipMemsetAsync(degbuf, 0, N_NODES * sizeof(float), stream);
    deg_kernel<<<(NEDGES + 255) / 256, 256, 0, stream>>>(ei, degbuf);
    invdeg_kernel<<<(N_NODES + 255) / 256, 256, 0, stream>>>(degbuf, invdbuf);
    prep_conv_w<<<(NLAYERS * KT * HID * HID + 255) / 256, 256, 0, stream>>>(tw, wtconv);
    prep_cat_w<<<(NLAYERS * 256 * HID + 255) / 256, 256, 0, stream>>>(wl, wr, wcat);

    const float* xcur = x_in;
    for (int i = 0; i < NLAYERS; ++i) {
        conv_wmma_kernel<<<(N_NODES + 63) / 64, 256, 0, stream>>>(
            xcur, wtconv + (size_t)i * KT * HID * HID, tb + i * HID, hbuf);
        (void)hipMemsetAsync(aggbuf, 0, (size_t)N_NODES * HID * sizeof(float), stream);
        long long sthreads = (long long)NEDGES * 32;
        scatter_kernel<<<(unsigned)((sthreads + 255) / 256), 256, 0, stream>>>(hbuf, ei, aggbuf);
        sage_wmma_kernel<<<(N_NODES + 31) / 32, 256, 0, stream>>>(
            aggbuf, invdbuf, hbuf, wcat + (size_t)i * 256 * HID, bl + i * HID,
            bng + i * HID, bnb + i * HID, bnm + i * HID, bnv + i * HID, xbuf);
        xcur = xbuf;
    }
    clf_kernel<<<N_NODES, 64, 0, stream>>>(xcur, clfw, clfb, out);
}